// SAGE_22162031247565
// MI455X (gfx1250) — compile-verified
//
#include <hip/hip_runtime.h>
#include <hip/hip_bf16.h>

#if defined(__HIP_DEVICE_COMPILE__) && !__has_builtin(__builtin_amdgcn_wmma_f32_16x16x4_f32)
#error "gfx1250 f32 WMMA builtin not available on this toolchain"
#endif

typedef __attribute__((ext_vector_type(2))) float v2f;
typedef __attribute__((ext_vector_type(8))) float v8f;

#define DIM 64
#define LDS_STRIDE 68   // 64 + 4 pad: 16-lane B-frag b64 loads hit 64 distinct banks

// ---------------------------------------------------------------------------
// degree: atomic count of incoming edges per node
__global__ __launch_bounds__(256) void sage_deg(const int* __restrict__ dst,
                                                float* __restrict__ deg, int nE) {
    int e = blockIdx.x * 256 + threadIdx.x;
    if (e < nE) atomicAdd(&deg[dst[e]], 1.0f);
}

// deg -> 1/max(deg,1) in place
__global__ __launch_bounds__(256) void sage_recip(float* __restrict__ deg, int n) {
    int i = blockIdx.x * 256 + threadIdx.x;
    if (i < n) deg[i] = 1.0f / fmaxf(deg[i], 1.0f);
}

// scatter-add: agg[dst] += feat[src].  16 threads/edge, float4 chunks.
__global__ __launch_bounds__(256) void sage_scatter(const float* __restrict__ feat,
                                                    const int* __restrict__ src,
                                                    const int* __restrict__ dst,
                                                    float* __restrict__ agg, int nE) {
    unsigned tid = blockIdx.x * 256u + threadIdx.x;
    int e = (int)(tid >> 4);
    if (e >= nE) return;
    int c = ((int)tid & 15) << 2;
    int s = src[e];
    int d = dst[e];
    const float4 v = *(const float4*)(feat + (size_t)s * DIM + c);
    float* o = agg + (size_t)d * DIM + c;
    atomicAdd(o + 0, v.x);
    atomicAdd(o + 1, v.y);
    atomicAdd(o + 2, v.z);
    atomicAdd(o + 3, v.w);
}

// ---------------------------------------------------------------------------
// Dense phase: out = relu(normalize((agg*deg_inv) @ Wl^T + bias + xin @ Wr^T))
// One wave per 16-row tile, 8 waves per block. K=64 via 16 x WMMA f32 16x16x4
// per (output-col-tile, matrix), 4 col tiles, 2 matrices -> 128 WMMA / wave.
__global__ __launch_bounds__(256) void sage_dense(const float* __restrict__ agg,
                                                  const float* __restrict__ xin,
                                                  const float* __restrict__ dinv,
                                                  const float* __restrict__ Wl,
                                                  const float* __restrict__ bias,
                                                  const float* __restrict__ Wr,
                                                  float* __restrict__ out,
                                                  int nTiles) {
    __shared__ float lwl[DIM * LDS_STRIDE];
    __shared__ float lwr[DIM * LDS_STRIDE];
    int t = threadIdx.x;
    // stage both 64x64 weight matrices into LDS (row-major, padded stride)
    for (int i = t; i < DIM * DIM; i += 256) {
        int r = i >> 6, c = i & 63;
        lwl[r * LDS_STRIDE + c] = Wl[i];
        lwr[r * LDS_STRIDE + c] = Wr[i];
    }
    __syncthreads();

    int wave = t >> 5;
    int lane = t & 31;
    int tile = blockIdx.x * 8 + wave;
    if (tile >= nTiles) return;
    int row0 = tile * 16;

    // 32-bit A-matrix 16x4 layout: lanes 0-15 hold K={0,1}, lanes 16-31 K={2,3}
    int m  = lane & 15;          // row within tile (A/X) or column within tile (B)
    int kb = (lane >> 4) << 1;   // K sub-offset for this lane half
    int row = row0 + m;

    float di = dinv[row];
    const float* arow = agg + (size_t)row * DIM;
    const float* xrow = xin + (size_t)row * DIM;

    v2f afrag[16], xfrag[16];
#pragma unroll
    for (int s = 0; s < 16; ++s) {
        int k = 4 * s + kb;
        v2f a;  a.x = arow[k]; a.y = arow[k + 1];
        afrag[s] = a * di;                       // fold deg_inv into A
        v2f xv; xv.x = xrow[k]; xv.y = xrow[k + 1];
        xfrag[s] = xv;
    }

    v8f acc[4];
#pragma unroll
    for (int j = 0; j < 4; ++j) {
        int nIdx = 16 * j + m;                   // output channel this lane feeds
        const float* blp = &lwl[nIdx * LDS_STRIDE];
        const float* brp = &lwr[nIdx * LDS_STRIDE];
        v8f c = {};
#pragma unroll
        for (int s = 0; s < 16; ++s) {
            int k = 4 * s + kb;
            v2f bf; bf.x = blp[k]; bf.y = blp[k + 1];   // B(k,n) = Wl[n,k]
            c = __builtin_amdgcn_wmma_f32_16x16x4_f32(
                    false, afrag[s], false, bf, (short)0, c, false, false);
            v2f br; br.x = brp[k]; br.y = brp[k + 1];   // B(k,n) = Wr[n,k]
            c = __builtin_amdgcn_wmma_f32_16x16x4_f32(
                    false, xfrag[s], false, br, (short)0, c, false, false);
        }
        acc[j] = c + bias[16 * j + m];           // bias depends only on column
    }

    // Row L2 norm: C/D layout -> VGPR v holds row M=v (lanes 0-15) / M=v+8
    // (lanes 16-31). Reduce sum-of-squares across the 16-lane half.
    int rbase = row0 + ((lane >> 4) << 3);
#pragma unroll
    for (int v = 0; v < 8; ++v) {
        float s = 0.0f;
#pragma unroll
        for (int j = 0; j < 4; ++j) { float h = acc[j][v]; s += h * h; }
        s += __shfl_xor(s, 1);
        s += __shfl_xor(s, 2);
        s += __shfl_xor(s, 4);
        s += __shfl_xor(s, 8);                   // stays within 16-lane half
        float inv = 1.0f / fmaxf(sqrtf(s), 1e-12f);
        float* orow = out + (size_t)(rbase + v) * DIM;
#pragma unroll
        for (int j = 0; j < 4; ++j)
            orow[16 * j + m] = fmaxf(acc[j][v], 0.0f) * inv;
    }
}

// ---------------------------------------------------------------------------
extern "C" void kernel_launch(void* const* d_in, const int* in_sizes, int n_in,
                              void* d_out, int out_size, void* d_ws, size_t ws_size,
                              hipStream_t stream) {
    const float* x   = (const float*)d_in[0];
    const int*   ei  = (const int*)d_in[1];      // [2, nE] int32
    const float* Wl0 = (const float*)d_in[2];
    const float* bl0 = (const float*)d_in[3];
    const float* Wr0 = (const float*)d_in[4];
    const float* Wl1 = (const float*)d_in[5];
    const float* bl1 = (const float*)d_in[6];
    const float* Wr1 = (const float*)d_in[7];

    const int nN = in_sizes[0] / DIM;            // 100000
    const int nE = in_sizes[1] / 2;              // 1600000
    const int* src = ei;
    const int* dst = ei + nE;

    float* deg = (float*)d_ws;                                   // nN floats
    size_t degBytes = ((size_t)nN * sizeof(float) + 255) & ~(size_t)255;
    float* agg = (float*)((char*)d_ws + degBytes);               // nN*64 floats
    size_t aggBytes = (size_t)nN * DIM * sizeof(float);
    float* out = (float*)d_out;

    const int nTiles = (nN + 15) / 16;                           // 6250
    dim3 denseGrid((nTiles + 7) / 8);
    dim3 scatGrid((unsigned)(((size_t)nE * 16 + 255) / 256));
    dim3 edgeGrid((nE + 255) / 256);
    dim3 nodeGrid((nN + 255) / 256);
    dim3 blk(256);

    // degree -> deg_inv (in place)
    (void)hipMemsetAsync(deg, 0, (size_t)nN * sizeof(float), stream);
    sage_deg<<<edgeGrid, blk, 0, stream>>>(dst, deg, nE);
    sage_recip<<<nodeGrid, blk, 0, stream>>>(deg, nN);

    // ---- layer 0: h0 -> d_out
    (void)hipMemsetAsync(agg, 0, aggBytes, stream);
    sage_scatter<<<scatGrid, blk, 0, stream>>>(x, src, dst, agg, nE);
    sage_dense<<<denseGrid, blk, 0, stream>>>(agg, x, deg, Wl0, bl0, Wr0, out, nTiles);

    // ---- layer 1: reads h0 from d_out, overwrites d_out in place
    (void)hipMemsetAsync(agg, 0, aggBytes, stream);
    sage_scatter<<<scatGrid, blk, 0, stream>>>(out, src, dst, agg, nE);
    sage_dense<<<denseGrid, blk, 0, stream>>>(agg, out, deg, Wl1, bl1, Wr1, out, nTiles);
}